// SparseGATLayer_66932770340994
// MI455X (gfx1250) — compile-verified
//
#include <hip/hip_runtime.h>
#include <hip/hip_bf16.h>

#define IN_DIM 128
#define HEADS 4
#define OUT_DIM 32
#define HD 128          // HEADS*OUT_DIM
#define NEG_SLOPE 0.2f

typedef __attribute__((ext_vector_type(2))) float v2f;
typedef __attribute__((ext_vector_type(8))) float v8f;

// ---- monotone float<->uint flip so atomicMax(u32) orders like float max ----
__device__ __forceinline__ unsigned flipF(float f) {
    unsigned u = __float_as_uint(f);
    return (u >> 31) ? ~u : (u | 0x80000000u);
}
__device__ __forceinline__ float unflipF(unsigned u) {
    return (u >> 31) ? __uint_as_float(u & 0x7FFFFFFFu) : __uint_as_float(~u);
}

// ============================================================================
// Kernel 1: x_proj = x @ W^T using V_WMMA_F32_16X16X4_F32.
// Block = 256 threads = 8 waves; each wave owns a 16-row tile of x_proj and
// all 8 column tiles (8 x v8f accumulators).
// W lives in LDS as K-pairs: sW2[p*128+n] = {W[n][2p], W[n][2p+1]} (64 KB),
// so every B fragment is ONE contiguous ds_load_b64 directly into an
// even-aligned VGPR pair -> no v_mov shuffling, and adjacent t-tiles (128 B
// apart) can merge into ds_load_2addr_b64. Lanes 0-15 sweep consecutive n,
// so the 8-byte accesses are bank-conflict-free.
// A-fragment 16x4 f32 layout: lanes 0-15 -> M=lane, K={0,1}; lanes 16-31 -> K={2,3}.
// ============================================================================
__global__ __launch_bounds__(256) void gat_gemm_wmma(
        const float* __restrict__ x, const float* __restrict__ W,
        float* __restrict__ xp, int n_nodes) {
    __shared__ v2f sW2[(IN_DIM / 2) * HD];   // 64 KB
    for (int j = threadIdx.x; j < (IN_DIM / 2) * HD; j += 256) {
        int p = j >> 7, n = j & 127;         // pair index, output column
        const float* wr = W + (size_t)n * IN_DIM + 2 * p;
        v2f w; w.x = wr[0]; w.y = wr[1];
        sW2[j] = w;                          // conflict-free LDS write
    }
    __syncthreads();

    const int wave = threadIdx.x >> 5;
    const int lane = threadIdx.x & 31;
    const int tile = blockIdx.x * 8 + wave;      // 16-row tile index
    if (tile * 16 >= n_nodes) return;            // wave-uniform: EXEC stays all-1s

    const int row0  = tile * 16;
    const int mlane = lane & 15;
    const int hi    = (lane < 16) ? 0 : 1;       // K offset = 2*hi

    v8f acc[8];
    v8f zero = {};
#pragma unroll
    for (int t = 0; t < 8; ++t) acc[t] = zero;

    int arow = row0 + mlane; if (arow >= n_nodes) arow = n_nodes - 1;
    const float* xrow = x + (size_t)arow * IN_DIM + 2 * hi;

    for (int kk = 0; kk < IN_DIM / 4; ++kk) {
        const v2f a = *(const v2f*)(xrow + kk * 4);          // 8B-aligned b64 load
        const v2f* brow = &sW2[(kk * 2 + hi) * HD + mlane];  // fragment base
#pragma unroll
        for (int t = 0; t < 8; ++t) {
            const v2f b = brow[t * 16];                      // single ds_load_b64
            acc[t] = __builtin_amdgcn_wmma_f32_16x16x4_f32(
                false, a, false, b, (short)0, acc[t], false, false);
        }
    }

    // C layout: VGPR r -> M = r (lanes 0-15) / M = r+8 (lanes 16-31), N = lane%16
    const int rbase = (lane < 16) ? 0 : 8;
#pragma unroll
    for (int t = 0; t < 8; ++t) {
#pragma unroll
        for (int r = 0; r < 8; ++r) {
            int row = row0 + rbase + r;
            if (row < n_nodes)
                xp[(size_t)row * HD + t * 16 + mlane] = acc[t][r];
        }
    }
}

// ============================================================================
// Kernel 2: per-node attention halves  a_src[n,h] = <x_proj[n,h,:], att_src[h,:]>
// One wave per node; lane d covers dim d of each head; 5-step xor reduction.
// ============================================================================
__global__ __launch_bounds__(256) void gat_scores(
        const float* __restrict__ xp,
        const float* __restrict__ att_src, const float* __restrict__ att_dst,
        float* __restrict__ a_src, float* __restrict__ a_dst, int n_nodes) {
    const int node = (int)((blockIdx.x * (size_t)blockDim.x + threadIdx.x) >> 5);
    const int lane = threadIdx.x & 31;
    if (node >= n_nodes) return;
    const float* row = xp + (size_t)node * HD;
#pragma unroll
    for (int h = 0; h < HEADS; ++h) {
        float v  = row[h * OUT_DIM + lane];
        float ps = v * att_src[h * OUT_DIM + lane];
        float pd = v * att_dst[h * OUT_DIM + lane];
#pragma unroll
        for (int off = 16; off > 0; off >>= 1) {
            ps += __shfl_xor(ps, off, 32);
            pd += __shfl_xor(pd, off, 32);
        }
        if (lane == 0) {
            a_src[node * HEADS + h] = ps;
            a_dst[node * HEADS + h] = pd;
        }
    }
}

// ============================================================================
// Kernel 3: init nodemax (flipped -1e15), denom=0, out=bias
// ============================================================================
__global__ void gat_init(unsigned* __restrict__ nodemax, float* __restrict__ denom,
                         float* __restrict__ out, const float* __restrict__ bias,
                         int n_nodes) {
    const int i = blockIdx.x * blockDim.x + threadIdx.x;
    if (i < n_nodes * HEADS) {
        nodemax[i] = flipF(-1e15f);
        denom[i]   = 0.0f;
    }
    if (i < n_nodes * HD) out[i] = bias[i & (HD - 1)];
}

// ============================================================================
// Kernel 4: per-edge logit + atomic segment-max (flipped u32 atomicMax in L2)
// ============================================================================
__global__ __launch_bounds__(256) void gat_edge_max(
        const int* __restrict__ src, const int* __restrict__ dst,
        const float* __restrict__ a_src, const float* __restrict__ a_dst,
        unsigned* __restrict__ nodemax, int n_edges) {
    const int e = blockIdx.x * blockDim.x + threadIdx.x;
    if (e >= n_edges) return;
    const int s = src[e], d = dst[e];
    const float4 as = *(const float4*)(a_src + (size_t)s * HEADS);
    const float4 ad = *(const float4*)(a_dst + (size_t)d * HEADS);
    float att[4] = {as.x + ad.x, as.y + ad.y, as.z + ad.z, as.w + ad.w};
#pragma unroll
    for (int h = 0; h < HEADS; ++h) {
        float v = att[h];
        v = (v >= 0.0f) ? v : NEG_SLOPE * v;
        atomicMax(&nodemax[(size_t)d * HEADS + h], flipF(v));
    }
}

// ============================================================================
// Kernel 5: denominator pass (recompute logit; exp; atomicAdd f32 in L2)
// ============================================================================
__global__ __launch_bounds__(256) void gat_edge_sum(
        const int* __restrict__ src, const int* __restrict__ dst,
        const float* __restrict__ a_src, const float* __restrict__ a_dst,
        const unsigned* __restrict__ nodemax, float* __restrict__ denom,
        int n_edges) {
    const int e = blockIdx.x * blockDim.x + threadIdx.x;
    if (e >= n_edges) return;
    const int s = src[e], d = dst[e];
    const float4 as = *(const float4*)(a_src + (size_t)s * HEADS);
    const float4 ad = *(const float4*)(a_dst + (size_t)d * HEADS);
    float att[4] = {as.x + ad.x, as.y + ad.y, as.z + ad.z, as.w + ad.w};
#pragma unroll
    for (int h = 0; h < HEADS; ++h) {
        float v = att[h];
        v = (v >= 0.0f) ? v : NEG_SLOPE * v;
        float m = unflipF(nodemax[(size_t)d * HEADS + h]);
        atomicAdd(&denom[(size_t)d * HEADS + h], __expf(v - m));
    }
}

// ============================================================================
// Kernel 6: weighted scatter-add aggregation. One wave per edge:
// lane covers 4 contiguous dims (float4 gather of x_proj[src], 4 f32 atomics
// into out[dst]); head h = lane>>3; softmax stats re-read (L1/L2 hits).
// ============================================================================
__global__ __launch_bounds__(256) void gat_aggregate(
        const int* __restrict__ src, const int* __restrict__ dst,
        const float* __restrict__ xp,
        const float* __restrict__ a_src, const float* __restrict__ a_dst,
        const unsigned* __restrict__ nodemax, const float* __restrict__ denom,
        float* __restrict__ out, int n_edges) {
    const int e    = (int)((blockIdx.x * (size_t)blockDim.x + threadIdx.x) >> 5);
    const int lane = threadIdx.x & 31;
    if (e >= n_edges) return;
    const int s = src[e], d = dst[e];
    const int h = lane >> 3;

    float att = a_src[(size_t)s * HEADS + h] + a_dst[(size_t)d * HEADS + h];
    att = (att >= 0.0f) ? att : NEG_SLOPE * att;
    const float m     = unflipF(nodemax[(size_t)d * HEADS + h]);
    const float den   = denom[(size_t)d * HEADS + h];
    const float alpha = __expf(att - m) / (den + 1e-12f);

    const int col = h * OUT_DIM + (lane & 7) * 4;
    const float4 v = *(const float4*)(xp + (size_t)s * HD + col);
    float* o = out + (size_t)d * HD + col;
    atomicAdd(o + 0, alpha * v.x);
    atomicAdd(o + 1, alpha * v.y);
    atomicAdd(o + 2, alpha * v.z);
    atomicAdd(o + 3, alpha * v.w);
}

// ============================================================================
extern "C" void kernel_launch(void* const* d_in, const int* in_sizes, int n_in,
                              void* d_out, int out_size, void* d_ws, size_t ws_size,
                              hipStream_t stream) {
    const float* x        = (const float*)d_in[0];
    const int*   ei       = (const int*)d_in[1];   // [2, E] flat
    const float* W        = (const float*)d_in[2];
    const float* att_src  = (const float*)d_in[3];
    const float* att_dst  = (const float*)d_in[4];
    const float* bias     = (const float*)d_in[5];
    float*       out      = (float*)d_out;

    const int n_nodes = in_sizes[0] / IN_DIM;
    const int n_edges = in_sizes[1] / 2;
    const int* srcp = ei;
    const int* dstp = ei + n_edges;

    // workspace carve-up (~57.6 MB total)
    float*    xp      = (float*)d_ws;                              // n_nodes*128
    float*    a_src   = xp + (size_t)n_nodes * HD;                 // n_nodes*4
    float*    a_dst   = a_src + (size_t)n_nodes * HEADS;           // n_nodes*4
    unsigned* nodemax = (unsigned*)(a_dst + (size_t)n_nodes * HEADS);
    float*    denom   = (float*)(nodemax + (size_t)n_nodes * HEADS);

    const int row_tiles = (n_nodes + 15) / 16;
    gat_gemm_wmma<<<(row_tiles + 7) / 8, 256, 0, stream>>>(x, W, xp, n_nodes);

    gat_scores<<<(int)(((size_t)n_nodes * 32 + 255) / 256), 256, 0, stream>>>(
        xp, att_src, att_dst, a_src, a_dst, n_nodes);

    gat_init<<<(n_nodes * HD + 255) / 256, 256, 0, stream>>>(
        nodemax, denom, out, bias, n_nodes);

    gat_edge_max<<<(n_edges + 255) / 256, 256, 0, stream>>>(
        srcp, dstp, a_src, a_dst, nodemax, n_edges);

    gat_edge_sum<<<(n_edges + 255) / 256, 256, 0, stream>>>(
        srcp, dstp, a_src, a_dst, nodemax, denom, n_edges);

    gat_aggregate<<<(int)(((size_t)n_edges * 32 + 255) / 256), 256, 0, stream>>>(
        srcp, dstp, xp, a_src, a_dst, nodemax, denom, out, n_edges);
}